// GraphConvolution_43791486550471
// MI455X (gfx1250) — compile-verified
//
#include <hip/hip_runtime.h>
#include <cstdint>

typedef float v2f __attribute__((ext_vector_type(2)));
typedef float v4f __attribute__((ext_vector_type(4)));
typedef float v8f __attribute__((ext_vector_type(8)));

#define IN_F 512
#define OUT_F 256
#define CHUNK_K 128
// K-pair-interleaved LDS layout: lds[(k>>1)*PSTRIDE + col*2 + (k&1)].
// PSTRIDE mod 64 == 32 -> the two half-waves of a b64 fragment read
// disjoint bank halves: conflict-free across all 64 LDS banks.
#define PSTRIDE 160

// ---------------------------------------------------------------------------
// Dense GEMM: support[N, 256] = input[N, 512] @ weight[512, 256]
// wave32 WMMA f32 16x16x4. Each wave: 16 rows x 64 cols (4 C tiles).
// Block: 8 waves -> 128 rows x 64 cols. gridDim.y = 256/64 = 4 col groups.
// Weight slab staged through LDS in K-chunks of 128, K-pair interleaved so
// each B fragment is a single aligned ds_load_b64 (no register shuffles).
// ---------------------------------------------------------------------------
__global__ __launch_bounds__(256) void gcn_gemm_wmma(
    const float* __restrict__ input, const float* __restrict__ weight,
    float* __restrict__ support, int nRows)
{
  __shared__ __align__(16) float ldsW[(CHUNK_K / 2) * PSTRIDE];   // 40 KB

  const int tid  = threadIdx.x;
  const int wave = tid >> 5;
  const int lane = tid & 31;
  const int m    = lane & 15;   // row-within-tile (A) / col-within-tile (B,C)
  const int hi   = lane >> 4;   // half-wave select

  const int rowTile = blockIdx.x * 8 + wave;
  const int row0    = rowTile * 16;
  const int col0    = blockIdx.y * 64;

  // Clamp so out-of-range waves still do (harmless) work and reach barriers.
  const bool storeOK = (row0 + 16) <= nRows;
  const int  aRow0   = storeOK ? row0 : (nRows - 16);

  // A fragment base: lane holds input[aRow0+m][k + 2*hi + {0,1}]  (f32 A layout)
  const float* aBase = input + (size_t)(aRow0 + m) * IN_F + 2 * hi;

  v8f acc0 = {}, acc1 = {}, acc2 = {}, acc3 = {};

  for (int kbase = 0; kbase < IN_F; kbase += CHUNK_K) {
    // Cooperative fill: item = (pair-row j, 4-col group). Two coalesced
    // global b128 loads (rows 2j, 2j+1), four aligned ds_store_b64.
    for (int i = tid; i < (CHUNK_K / 2) * 16; i += 256) {
      const int j  = i >> 4;        // 0..63 pair-row
      const int c  = (i & 15) * 4;  // col within slab
      const v4f w0 = *(const v4f*)(weight + (size_t)(kbase + 2 * j)     * OUT_F + col0 + c);
      const v4f w1 = *(const v4f*)(weight + (size_t)(kbase + 2 * j + 1) * OUT_F + col0 + c);
#pragma unroll
      for (int q = 0; q < 4; ++q) {
        v2f p; p.x = w0[q]; p.y = w1[q];
        *(v2f*)&ldsW[j * PSTRIDE + (c + q) * 2] = p;
      }
    }
    __syncthreads();

    // Warm the next A chunk into cache.
    if (kbase + CHUNK_K < IN_F)
      __builtin_prefetch(aBase + kbase + CHUNK_K, 0, 1);

#pragma unroll 4
    for (int kk = 0; kk < CHUNK_K; kk += 4) {
      const v2f a = *(const v2f*)(aBase + kbase + kk);     // K = kk+2hi, kk+2hi+1
      const int bRow = ((kk >> 1) + hi) * PSTRIDE;         // pair-row for this half-wave

      {
        const v2f b = *(const v2f*)&ldsW[bRow + (0 * 16 + m) * 2];
        acc0 = __builtin_amdgcn_wmma_f32_16x16x4_f32(false, a, false, b, (short)0, acc0, false, false);
      }
      {
        const v2f b = *(const v2f*)&ldsW[bRow + (1 * 16 + m) * 2];
        acc1 = __builtin_amdgcn_wmma_f32_16x16x4_f32(false, a, false, b, (short)0, acc1, false, false);
      }
      {
        const v2f b = *(const v2f*)&ldsW[bRow + (2 * 16 + m) * 2];
        acc2 = __builtin_amdgcn_wmma_f32_16x16x4_f32(false, a, false, b, (short)0, acc2, false, false);
      }
      {
        const v2f b = *(const v2f*)&ldsW[bRow + (3 * 16 + m) * 2];
        acc3 = __builtin_amdgcn_wmma_f32_16x16x4_f32(false, a, false, b, (short)0, acc3, false, false);
      }
    }
    __syncthreads();
  }

  if (storeOK) {
    // C layout: VGPR r -> row (r + 8*hi), col = m (16x16 f32 C/D layout)
#pragma unroll
    for (int r = 0; r < 8; ++r) {
      const size_t rowOff = (size_t)(row0 + hi * 8 + r) * OUT_F;
      support[rowOff + col0 +  0 + m] = acc0[r];
      support[rowOff + col0 + 16 + m] = acc1[r];
      support[rowOff + col0 + 32 + m] = acc2[r];
      support[rowOff + col0 + 48 + m] = acc3[r];
    }
  }
}

// ---------------------------------------------------------------------------
// SpMM scatter: out[rows[e]] += vals[e] * support[cols[e]]
// One wave per edge (32 lanes x 8 features). Edge indices are forced scalar
// (readfirstlane) so rows/cols/vals become uniform s_loads; gathers are two
// b128 loads per lane from the L2-resident support buffer; scatter uses
// native fp32 global atomics at device scope.
// ---------------------------------------------------------------------------
__global__ __launch_bounds__(256) void gcn_spmm_atomic(
    const int* __restrict__ adj_rows, const int* __restrict__ adj_cols,
    const float* __restrict__ adj_vals, const float* __restrict__ out_dummy_,
    const float* __restrict__ support, float* __restrict__ out, int nEdges)
{
  const int tid  = threadIdx.x;
  const int lane = tid & 31;
  int e = blockIdx.x * 8 + (tid >> 5);
  e = __builtin_amdgcn_readfirstlane(e);      // scalarize: uniform per wave
  if (e >= nEdges) return;

  const int   r = adj_rows[e];                // uniform -> s_load
  const int   c = adj_cols[e];                // uniform -> s_load
  const float v = adj_vals[e];                // uniform -> s_load

  const int f = lane * 8;
  const float* sp = support + (size_t)c * OUT_F + f;
  const v4f s0 = *(const v4f*)sp;
  const v4f s1 = *(const v4f*)(sp + 4);

  const float d0 = v * s0.x, d1 = v * s0.y, d2 = v * s0.z, d3 = v * s0.w;
  const float d4 = v * s1.x, d5 = v * s1.y, d6 = v * s1.z, d7 = v * s1.w;

  const uint64_t addr = (uint64_t)(uintptr_t)(out + (size_t)r * OUT_F + f);
  asm volatile("global_atomic_add_f32 %0, %1, off scope:SCOPE_DEV"           :: "v"(addr), "v"(d0) : "memory");
  asm volatile("global_atomic_add_f32 %0, %1, off offset:4 scope:SCOPE_DEV"  :: "v"(addr), "v"(d1) : "memory");
  asm volatile("global_atomic_add_f32 %0, %1, off offset:8 scope:SCOPE_DEV"  :: "v"(addr), "v"(d2) : "memory");
  asm volatile("global_atomic_add_f32 %0, %1, off offset:12 scope:SCOPE_DEV" :: "v"(addr), "v"(d3) : "memory");
  asm volatile("global_atomic_add_f32 %0, %1, off offset:16 scope:SCOPE_DEV" :: "v"(addr), "v"(d4) : "memory");
  asm volatile("global_atomic_add_f32 %0, %1, off offset:20 scope:SCOPE_DEV" :: "v"(addr), "v"(d5) : "memory");
  asm volatile("global_atomic_add_f32 %0, %1, off offset:24 scope:SCOPE_DEV" :: "v"(addr), "v"(d6) : "memory");
  asm volatile("global_atomic_add_f32 %0, %1, off offset:28 scope:SCOPE_DEV" :: "v"(addr), "v"(d7) : "memory");
}

// ---------------------------------------------------------------------------
// ReLU in-place over out (float4 granularity; out_size is a multiple of 4).
// ---------------------------------------------------------------------------
__global__ __launch_bounds__(256) void gcn_relu(float* __restrict__ out, int n4)
{
  const int i = blockIdx.x * 256 + threadIdx.x;
  if (i >= n4) return;
  v4f* p = (v4f*)out + i;
  v4f x = *p;
  x.x = fmaxf(x.x, 0.0f);
  x.y = fmaxf(x.y, 0.0f);
  x.z = fmaxf(x.z, 0.0f);
  x.w = fmaxf(x.w, 0.0f);
  *p = x;
}

extern "C" void kernel_launch(void* const* d_in, const int* in_sizes, int n_in,
                              void* d_out, int out_size, void* d_ws, size_t ws_size,
                              hipStream_t stream)
{
  const float* input    = (const float*)d_in[0];
  const float* weight   = (const float*)d_in[1];
  const int*   adj_rows = (const int*)d_in[2];
  const int*   adj_cols = (const int*)d_in[3];
  const float* adj_vals = (const float*)d_in[4];
  float* out     = (float*)d_out;
  float* support = (float*)d_ws;          // N * 256 fp32 scratch (~102.4 MB)

  const int nRows  = in_sizes[0] / IN_F;  // 100000
  const int nEdges = in_sizes[2];         // 3200000

  // d_out is poisoned by the harness; atomics need a zeroed accumulator.
  hipMemsetAsync(d_out, 0, (size_t)out_size * sizeof(float), stream);

  // 1) support = input @ weight   (WMMA f32)
  const int nTiles = (nRows + 15) / 16;
  dim3 ggrid((nTiles + 7) / 8, OUT_F / 64);
  gcn_gemm_wmma<<<ggrid, dim3(256), 0, stream>>>(input, weight, support, nRows);

  // 2) edge gather + scaled scatter-add (one wave per edge)
  const int eBlocks = (nEdges + 7) / 8;
  gcn_spmm_atomic<<<dim3(eBlocks), dim3(256), 0, stream>>>(
      adj_rows, adj_cols, adj_vals, nullptr, support, out, nEdges);

  // 3) ReLU
  const int n4 = out_size / 4;
  gcn_relu<<<dim3((n4 + 255) / 256), dim3(256), 0, stream>>>(out, n4);
}